// YoloLossLayer_55259049230932
// MI455X (gfx1250) — compile-verified
//
#include <hip/hip_runtime.h>
#include <cstdint>
#include <cstddef>

typedef __attribute__((ext_vector_type(2))) float v2f;
typedef __attribute__((ext_vector_type(8))) float v8f;

#define NCLS  80
#define ATTRS 85
#define MAXB  128

// Workspace layout (byte offsets into d_ws):
//   [0,   96)          int   counts[3*8]
//   [256, 256+49152)   float boxes[3][8][128][4]
//   [49408, ...)       float partials[NB][4]   (NB = 720 blocks)

__global__ void k_zero_counts(int* counts) {
  if (threadIdx.x < 24) counts[threadIdx.x] = 0;
}

// Pass A: compact positive GT boxes per (scale, batch).
__global__ void k_collect(const float* __restrict__ yt, int* __restrict__ counts,
                          float* __restrict__ boxes, int G, int s) {
  const int ncell = G * G * 3;
  const long long idx = (long long)blockIdx.x * blockDim.x + threadIdx.x;
  if (idx >= (long long)8 * ncell) return;
  const int b = (int)(idx / ncell);
  const int cell = (int)(idx % ncell);
  const size_t base = ((size_t)b * ncell + (size_t)cell) * ATTRS;
  const float obj = yt[base + 4];
  if (obj > 0.5f) {
    int pos = atomicAdd(&counts[s * 8 + b], 1);
    if (pos < MAXB) {
      float* dst = boxes + (((size_t)(s * 8 + b)) * MAXB + (size_t)pos) * 4;
      dst[0] = yt[base + 0];
      dst[1] = yt[base + 1];
      dst[2] = yt[base + 2];
      dst[3] = yt[base + 3];
    }
  }
}

__device__ __forceinline__ float log_clip(float v) {
  if (v == 0.0f) v = 1.0f;
  v = fminf(fmaxf(v, 1e-9f), 1e9f);
  return logf(v);
}

// Pass B: per-cell loss terms + deterministic block reduction to partials.
__global__ void __launch_bounds__(256)
k_main(const float* __restrict__ fm, const float* __restrict__ yt,
       const float* __restrict__ anchors, const int* __restrict__ counts,
       const float* __restrict__ boxes, float* __restrict__ partials,
       int G, int s, int bpb, int pbase) {
  __shared__ float sbox[MAXB * 4];   // staged GT boxes for this batch
  __shared__ float sred[256 * 4];    // block reduction scratch

  const int tid  = threadIdx.x;
  const int b    = blockIdx.x / bpb;
  const int tile = blockIdx.x % bpb;
  const int ncell = G * G * 3;
  const int cell  = tile * 256 + tid;
  const float stride = 608.0f / (float)G;
  const int s8b = s * 8 + b;
  const float* gb = boxes + (size_t)s8b * MAXB * 4;

  int cnt = counts[s8b];
  cnt = cnt > MAXB ? MAXB : cnt;

#if defined(__HIP_DEVICE_COMPILE__) && defined(__gfx1250__)
  // CDNA5 async copy: global -> LDS, 16B per lane, tracked with ASYNCcnt.
  if (tid < MAXB) {
    unsigned lds_addr = (unsigned)(unsigned long long)(uintptr_t)&sbox[tid * 4];
    const float* gsrc = gb + (size_t)tid * 4;
    asm volatile("global_load_async_to_lds_b128 %0, %1, off"
                 :: "v"(lds_addr), "v"(gsrc)
                 : "memory");
  }
  asm volatile("s_wait_asynccnt 0" ::: "memory");
#else
  if (tid < MAXB) {
    sbox[tid * 4 + 0] = gb[tid * 4 + 0];
    sbox[tid * 4 + 1] = gb[tid * 4 + 1];
    sbox[tid * 4 + 2] = gb[tid * 4 + 2];
    sbox[tid * 4 + 3] = gb[tid * 4 + 3];
  }
#endif
  __syncthreads();

  float lxy = 0.f, lwh = 0.f, lcf = 0.f, lpb = 0.f;
  if (cell < ncell && b < 8) {
    const int a   = cell % 3;
    const int xyi = cell / 3;
    const int gx  = xyi % G;
    const int gy  = xyi / G;
    const size_t GG = (size_t)G * G;
    const size_t fmbase = ((size_t)b * (3 * ATTRS) + (size_t)a * ATTRS) * GG +
                          (size_t)gy * G + (size_t)gx;
    const size_t ytbase = ((size_t)b * ncell + (size_t)cell) * ATTRS;

    const float tx = fm[fmbase + 0 * GG];
    const float ty = fm[fmbase + 1 * GG];
    const float tw = fm[fmbase + 2 * GG];
    const float th = fm[fmbase + 3 * GG];
    const float tc = fm[fmbase + 4 * GG];
    const float obj = yt[ytbase + 4];

    const int ab = (6 - 3 * s + a) * 2;  // fm0 -> anchors[6:9], fm1 -> [3:6], fm2 -> [0:3]
    const float aw = anchors[ab + 0];
    const float ah = anchors[ab + 1];

    const float sx = 1.f / (1.f + expf(-tx));
    const float sy = 1.f / (1.f + expf(-ty));
    const float pw = expf(tw) * aw;
    const float ph = expf(th) * ah;
    const float pcx = (sx + (float)gx) * stride;
    const float pcy = (sy + (float)gy) * stride;

    // ignore mask: max IoU against positive GT boxes of this batch
    float best = 0.f;
    const float pa = pw * ph;
    const float pminx = pcx - pw * 0.5f, pmaxx = pcx + pw * 0.5f;
    const float pminy = pcy - ph * 0.5f, pmaxy = pcy + ph * 0.5f;
    for (int i = 0; i < cnt; ++i) {
      const float bx = sbox[i * 4 + 0], by = sbox[i * 4 + 1];
      const float bw = sbox[i * 4 + 2], bh = sbox[i * 4 + 3];
      const float ix = fminf(pmaxx, bx + bw * 0.5f) - fmaxf(pminx, bx - bw * 0.5f);
      const float iy = fminf(pmaxy, by + bh * 0.5f) - fmaxf(pminy, by - bh * 0.5f);
      const float iw = fmaxf(ix, 0.f), ih = fmaxf(iy, 0.f);
      const float inter = iw * ih;
      const float iou = inter / (pa + bw * bh - inter + 1e-10f);
      best = fmaxf(best, iou);
    }
    const float ign = (best <= 0.5f) ? 1.f : 0.f;

    // confidence (focal BCE) — applies to every cell
    const float bce = fmaxf(tc, 0.f) - tc * obj + log1pf(expf(-fabsf(tc)));
    const float sc  = 1.f / (1.f + expf(-tc));
    const float fd  = obj - sc;
    lcf = (obj * bce + 0.5f * (1.f - obj) * ign * bce) * (fd * fd);

    if (obj > 0.f) {  // rare path (~0.3% of cells)
      const float yx = yt[ytbase + 0], yy = yt[ytbase + 1];
      const float yw = yt[ytbase + 2], yh = yt[ytbase + 3];
      const float bscale = 2.f - (yw / 608.f) * (yh / 608.f);
      const float dx = sx - (yx / stride - (float)gx);
      const float dy = sy - (yy / stride - (float)gy);
      lxy = 5.f * obj * bscale * (dx * dx + dy * dy);

      const float ptw = log_clip(pw / aw);
      const float pth = log_clip(ph / ah);
      const float ttw = log_clip(yw / aw);
      const float tth = log_clip(yh / ah);
      const float dw = ptw - ttw, dh = pth - tth;
      lwh = 5.f * obj * bscale * (dw * dw + dh * dh);

      __builtin_prefetch(fm + fmbase + 5 * GG, 0, 1);
      float cls = 0.f;
      for (int c = 0; c < NCLS; ++c) {
        const float pl  = fm[fmbase + (size_t)(5 + c) * GG];
        const float tgt = 0.99f * yt[ytbase + 5 + c] + (0.01f / 80.0f);
        cls += fmaxf(pl, 0.f) - pl * tgt + log1pf(expf(-fabsf(pl)));
      }
      lpb = obj * cls;
    }
  }

  // deterministic in-block tree reduction of (lxy, lwh, lcf, lpb)
  sred[tid * 4 + 0] = lxy;
  sred[tid * 4 + 1] = lwh;
  sred[tid * 4 + 2] = lcf;
  sred[tid * 4 + 3] = lpb;
  __syncthreads();
  for (int sft = 128; sft > 0; sft >>= 1) {
    if (tid < sft) {
      sred[tid * 4 + 0] += sred[(tid + sft) * 4 + 0];
      sred[tid * 4 + 1] += sred[(tid + sft) * 4 + 1];
      sred[tid * 4 + 2] += sred[(tid + sft) * 4 + 2];
      sred[tid * 4 + 3] += sred[(tid + sft) * 4 + 3];
    }
    __syncthreads();
  }
  if (tid == 0) {
    float* p = partials + (size_t)(pbase + blockIdx.x) * 4;
    p[0] = sred[0]; p[1] = sred[1]; p[2] = sred[2]; p[3] = sred[3];
  }
}

// Final reduction: one wave32. Uses V_WMMA_F32_16X16X4_F32 (full f32 precision)
// with A = ones(16x4): D[m,n] = sum_k B[k,n] + C[m,n]. We pack 16 partial
// 4-vectors per step: component = n%4, group = n/4, index-in-group = k.
__global__ void k_reduce(const float* __restrict__ partials, int nb,
                         float* __restrict__ out) {
  const int lane = threadIdx.x;            // blockDim.x == 32 (EXEC all ones)
  const int n    = lane & 15;
  const int kb   = (lane < 16) ? 0 : 2;    // f32 B 4x16: lanes 0-15 hold K0/K1, 16-31 hold K2/K3
  const int i0   = ((n >> 2) << 2) + kb;   // partial index within group of 16
  const int comp = n & 3;
  const int nb16 = (nb / 16) * 16;
  float acc0 = 0.f;

#if defined(__HIP_DEVICE_COMPILE__) && __has_builtin(__builtin_amdgcn_wmma_f32_16x16x4_f32)
  v2f amat = {1.0f, 1.0f};
  v8f cacc = {0.f, 0.f, 0.f, 0.f, 0.f, 0.f, 0.f, 0.f};
  for (int off = 0; off < nb16; off += 16) {
    v2f bmat;
    bmat.x = partials[(size_t)(off + i0) * 4 + comp];
    bmat.y = partials[(size_t)(off + i0 + 1) * 4 + comp];
    cacc = __builtin_amdgcn_wmma_f32_16x16x4_f32(
        /*neg_a=*/false, amat, /*neg_b=*/false, bmat,
        /*c_mod=*/(short)0, cacc, /*reuse_a=*/false, /*reuse_b=*/false);
  }
  acc0 = cacc[0];  // lane n (0..15): D[0,n] = column sum over all K
#else
  for (int off = 0; off < nb16; off += 16) {
    const int g = (n >> 2) << 2;
    acc0 += partials[(size_t)(off + g + 0) * 4 + comp] +
            partials[(size_t)(off + g + 1) * 4 + comp] +
            partials[(size_t)(off + g + 2) * 4 + comp] +
            partials[(size_t)(off + g + 3) * 4 + comp];
    if (lane >= 16) acc0 = 0.f;  // avoid double count in fallback
  }
#endif

  float sums[4];
#pragma unroll
  for (int c = 0; c < 4; ++c) {
    sums[c] = __shfl(acc0, c, 32) + __shfl(acc0, c + 4, 32) +
              __shfl(acc0, c + 8, 32) + __shfl(acc0, c + 12, 32);
  }
  if (lane == 0) {
    for (int i = nb16; i < nb; ++i)
      for (int c = 0; c < 4; ++c) sums[c] += partials[(size_t)i * 4 + c];
    const float invB = 1.0f / 8.0f;  // sum(loss)/BATCH per scale == total/8
    const float xy = sums[0] * invB, wh = sums[1] * invB;
    const float cf = sums[2] * invB, pb = sums[3] * invB;
    out[0] = xy + wh + cf + pb;
    out[1] = xy; out[2] = wh; out[3] = cf; out[4] = pb;
  }
}

extern "C" void kernel_launch(void* const* d_in, const int* in_sizes, int n_in,
                              void* d_out, int out_size, void* d_ws, size_t ws_size,
                              hipStream_t stream) {
  (void)in_sizes; (void)n_in; (void)out_size; (void)ws_size;
  const float* fm[3] = {(const float*)d_in[0], (const float*)d_in[1], (const float*)d_in[2]};
  const float* yt[3] = {(const float*)d_in[3], (const float*)d_in[4], (const float*)d_in[5]};
  const float* anchors = (const float*)d_in[6];

  int*   counts   = (int*)d_ws;
  float* boxes    = (float*)((char*)d_ws + 256);
  float* partials = (float*)((char*)d_ws + 256 + (size_t)3 * 8 * MAXB * 4 * sizeof(float));

  const int G[3] = {19, 38, 76};

  k_zero_counts<<<1, 32, 0, stream>>>(counts);

  for (int s = 0; s < 3; ++s) {
    const int total = 8 * G[s] * G[s] * 3;
    k_collect<<<(total + 255) / 256, 256, 0, stream>>>(yt[s], counts, boxes, G[s], s);
  }

  int pbase = 0;
  for (int s = 0; s < 3; ++s) {
    const int ncell = G[s] * G[s] * 3;
    const int bpb = (ncell + 255) / 256;  // blocks per batch image
    k_main<<<bpb * 8, 256, 0, stream>>>(fm[s], yt[s], anchors, counts, boxes,
                                        partials, G[s], s, bpb, pbase);
    pbase += bpb * 8;
  }

  k_reduce<<<1, 32, 0, stream>>>(partials, pbase, (float*)d_out);
}